// MeshNN_71889162600997
// MI455X (gfx1250) — compile-verified
//
#include <hip/hip_runtime.h>
#include <cstdint>

#define BLOCK 256

// ---------------------------------------------------------------------------
// Pre-pass: transpose nodal_values (2, n_nodes) SoA -> (n_nodes) float2 AoS
// so the hot per-point gather is a single b64 per node instead of two b32s.
// ---------------------------------------------------------------------------
__global__ __launch_bounds__(BLOCK) void pack_nodal_kernel(
    const float* __restrict__ nv, float2* __restrict__ nv2, int n_nodes)
{
    int i = blockIdx.x * BLOCK + threadIdx.x;
    if (i < n_nodes) {
        float2 v;
        v.x = nv[i];
        v.y = nv[n_nodes + i];
        nv2[i] = v;
    }
}

// ---------------------------------------------------------------------------
// Main kernel: one thread per point. T6 quadratic shape functions, faithful
// to the reference (including its quirky d1/d2/d3 usage), then a length-6
// dot per component.
//
// Issue order (critical-path first):
//   cell_id -> conn row -> corner-coords gathers (LOADcnt, needed by math)
//   -> 6 async nodal gathers to LDS (ASYNCcnt, needed only at the end)
//   -> barycentric math (overlaps both) -> s_wait_asynccnt -> LDS dot -> NT store
// ---------------------------------------------------------------------------
template <bool PACKED>
__global__ __launch_bounds__(BLOCK) void interp_t6_kernel(
    const float2* __restrict__ x,        // (n_points) xy
    const float2* __restrict__ coords,   // (n_nodes) xy
    const float*  __restrict__ nv,       // (2, n_nodes) original layout
    const float2* __restrict__ nv2,      // (n_nodes) packed (PACKED only)
    const int*    __restrict__ cell_id,  // (n_points)
    const int*    __restrict__ conn,     // (n_cells, 6), 1-based node ids
    float*        __restrict__ out,      // (2, n_points)
    int n_points, int n_nodes)
{
    // [k][tid] layout: b64 LDS reads across a wave walk consecutive 8B slots
    // -> conflict-free; backend fuses k-planes into ds_load_2addr_stride64_b64.
    __shared__ float2 s_nv[6][BLOCK];

    int n = blockIdx.x * BLOCK + threadIdx.x;
    if (n >= n_points) return;

    // Streaming inputs: non-temporal so they don't evict the L2-resident
    // gather tables (coords / conn / nodal fit easily in the 192MB L2).
    int cell = __builtin_nontemporal_load(cell_id + n);
    double xraw = __builtin_nontemporal_load((const double*)x + n);
    float2 xy;
    __builtin_memcpy(&xy, &xraw, 8);

    // Connectivity row gather: 24B row, 8B aligned -> 3 x b64.
    const int* crow = conn + cell * 6;
    int2 c01 = *(const int2*)(crow + 0);
    int2 c23 = *(const int2*)(crow + 2);
    int2 c45 = *(const int2*)(crow + 4);
    int i0 = c01.x - 1, i1 = c01.y - 1, i2 = c23.x - 1;
    int i3 = c23.y - 1, i4 = c45.x - 1, i5 = c45.y - 1;

    // Corner-node coordinate gathers FIRST: they feed the math directly and
    // the asm memory clobbers below would otherwise pin them behind the
    // async issues.
    float2 p1 = coords[i0];
    float2 p2 = coords[i1];
    float2 p3 = coords[i2];

    if constexpr (PACKED) {
        // CDNA5 async global->LDS gathers (ASYNCcnt). These complete in the
        // background while the wave does the barycentric math below; we only
        // wait right before the final dot product.
        int idx6[6] = {i0, i1, i2, i3, i4, i5};
#pragma unroll
        for (int k = 0; k < 6; ++k) {
            // Generic LDS pointer: low 32 bits are the wave-relative LDS
            // byte address (aperture bits live in the high dword).
            uint32_t la = (uint32_t)(uintptr_t)(&s_nv[k][threadIdx.x]);
            uint64_t ga = (uint64_t)(uintptr_t)(nv2 + idx6[k]);
            asm volatile("global_load_async_to_lds_b64 %0, %1, off"
                         :: "v"(la), "v"(ga)
                         : "memory");
        }
    }

    float x1 = p1.x, y1 = p1.y;
    float x2 = p2.x, y2 = p2.y;
    float x3 = p3.x, y3 = p3.y;
    float xp = xy.x, yp = xy.y;

    // Faithful transcription of the reference (d2 == d1, d3 == -d1
    // mathematically, but keep the exact expressions / rounding pattern).
    float d1 = x1 * (y3 - y2) + x2 * (y1 - y3) + x3 * (y2 - y1);
    float d2 = -x1 * y2 + x1 * y3 + x2 * y1 - x2 * y3 - x3 * y1 + x3 * y2;
    float d3 = x1 * (y2 - y3) + x2 * (y3 - y1) + x3 * (y1 - y2);
    float r1 = 1.0f / d1;
    float r2 = 1.0f / d2;
    float r3 = 1.0f / d3;

    float l1 = xp * (y3 - y2) * r1 + yp * (x2 - x3) * r2 + (x3 * y2 - x2 * y3) * r2;
    float l2 = xp * (y1 - y3) * r2 + yp * (x1 - x3) * r3 + (x3 * y1 - x1 * y3) * r3;
    float l3 = xp * (y1 - y2) * r3 + yp * (x1 - x2) * r2 + (x2 * y1 - x1 * y2) * r2;

    float N1 = l1 * (2.0f * l1 - 1.0f);
    float N2 = l2 * (2.0f * l2 - 1.0f);
    float N3 = l3 * (2.0f * l3 - 1.0f);
    float N4 = 4.0f * l1 * l2;
    float N5 = 4.0f * l2 * l3;
    float N6 = 4.0f * l3 * l1;

    float2 v0, v1, v2, v3, v4, v5;
    if constexpr (PACKED) {
        // Drain the async gathers, then conflict-free LDS reads
        // (fused to ds_load_2addr_stride64_b64 by the backend).
        asm volatile("s_wait_asynccnt 0x0" ::: "memory");
        v0 = s_nv[0][threadIdx.x];
        v1 = s_nv[1][threadIdx.x];
        v2 = s_nv[2][threadIdx.x];
        v3 = s_nv[3][threadIdx.x];
        v4 = s_nv[4][threadIdx.x];
        v5 = s_nv[5][threadIdx.x];
    } else {
        // Fallback: 12 x b32 gathers from the original (2, n_nodes) layout.
        v0.x = nv[i0]; v0.y = nv[n_nodes + i0];
        v1.x = nv[i1]; v1.y = nv[n_nodes + i1];
        v2.x = nv[i2]; v2.y = nv[n_nodes + i2];
        v3.x = nv[i3]; v3.y = nv[n_nodes + i3];
        v4.x = nv[i4]; v4.y = nv[n_nodes + i4];
        v5.x = nv[i5]; v5.y = nv[n_nodes + i5];
    }

    float o0 = N1 * v0.x + N2 * v1.x + N3 * v2.x
             + N4 * v3.x + N5 * v4.x + N6 * v5.x;
    float o1 = N1 * v0.y + N2 * v1.y + N3 * v2.y
             + N4 * v3.y + N5 * v4.y + N6 * v5.y;

    // Write-once output: non-temporal stores.
    __builtin_nontemporal_store(o0, out + n);
    __builtin_nontemporal_store(o1, out + n_points + n);
}

// ---------------------------------------------------------------------------
// Host-side launcher.
// ---------------------------------------------------------------------------
extern "C" void kernel_launch(void* const* d_in, const int* in_sizes, int n_in,
                              void* d_out, int out_size, void* d_ws, size_t ws_size,
                              hipStream_t stream)
{
    const float2* x      = (const float2*)d_in[0];  // (N_POINTS, 2) f32
    const float2* coords = (const float2*)d_in[1];  // (N_NODES, 2)  f32
    const float*  nv     = (const float*) d_in[2];  // (2, N_NODES)  f32
    const int*    cellid = (const int*)   d_in[3];  // (N_POINTS)    i32
    const int*    conn   = (const int*)   d_in[4];  // (N_CELLS, 6)  i32
    float* out = (float*)d_out;                     // (2, N_POINTS) f32

    int n_points = in_sizes[0] / 2;
    int n_nodes  = in_sizes[1] / 2;

    int grid = (n_points + BLOCK - 1) / BLOCK;
    bool packed = ws_size >= (size_t)n_nodes * sizeof(float2);

    if (packed) {
        float2* nv2 = (float2*)d_ws;
        int pgrid = (n_nodes + BLOCK - 1) / BLOCK;
        pack_nodal_kernel<<<pgrid, BLOCK, 0, stream>>>(nv, nv2, n_nodes);
        interp_t6_kernel<true><<<grid, BLOCK, 0, stream>>>(
            x, coords, nv, nv2, cellid, conn, out, n_points, n_nodes);
    } else {
        interp_t6_kernel<false><<<grid, BLOCK, 0, stream>>>(
            x, coords, nv, nullptr, cellid, conn, out, n_points, n_nodes);
    }
}